// UpDown_87737591923202
// MI455X (gfx1250) — compile-verified
//
#include <hip/hip_runtime.h>
#include <hip/hip_bf16.h>
#include <math.h>

typedef __attribute__((ext_vector_type(2))) float v2f;
typedef __attribute__((ext_vector_type(8))) float v8f;

#define N_REG   576
#define IMG     2048
#define EMBED   1024
#define H1      1024
#define H2      1024
#define MID     512
#define VOCAB   32000
#define IN1     4096   // H2 + IMG + EMBED
#define IN2     3072   // IMG + H1

__device__ __forceinline__ float sigm(float x) { return 1.0f / (1.0f + __expf(-x)); }

// ---------------------------------------------------------------------------
// x1 = [h2 (1024) | mean(V,axis=0) (2048) | emb[w] (1024)]
// ---------------------------------------------------------------------------
__global__ void build_x1(const float* __restrict__ V, const float* __restrict__ h2,
                         const float* __restrict__ emb, const int* __restrict__ w,
                         float* __restrict__ x1) {
    int j = blockIdx.x * blockDim.x + threadIdx.x;
    if (j >= IN1) return;
    if (j < H2) {
        x1[j] = h2[j];
    } else if (j < H2 + IMG) {
        int col = j - H2;
        float s = 0.0f;
        for (int r = 0; r < N_REG; ++r) s += V[(size_t)r * IMG + col];
        x1[j] = s * (1.0f / (float)N_REG);
    } else {
        x1[j] = emb[(size_t)w[0] * EMBED + (j - H2 - IMG)];
    }
}

// ---------------------------------------------------------------------------
// Generic GEMV with bias: out[r] = dot(W[r,:n], x) + b[r].  One wave per row,
// float4 (global_load_b128) coalesced streaming — this path is HBM-bound.
// ---------------------------------------------------------------------------
__global__ void gemv_bias(const float* __restrict__ W, const float* __restrict__ x,
                          const float* __restrict__ b, float* __restrict__ out,
                          int rows, int n) {
    int wave = (blockIdx.x * blockDim.x + threadIdx.x) >> 5;
    int lane = threadIdx.x & 31;
    if (wave >= rows) return;
    const float4* wr = (const float4*)(W + (size_t)wave * n);
    const float4* xr = (const float4*)x;
    int nv = n >> 2;
    float acc = 0.0f;
    for (int i = lane; i < nv; i += 32) {
        float4 wv = wr[i], xv = xr[i];
        acc = fmaf(wv.x, xv.x, acc); acc = fmaf(wv.y, xv.y, acc);
        acc = fmaf(wv.z, xv.z, acc); acc = fmaf(wv.w, xv.w, acc);
    }
#pragma unroll
    for (int off = 16; off > 0; off >>= 1) acc += __shfl_down(acc, off, 32);
    if (lane == 0) out[wave] = acc + b[wave];
}

// ---------------------------------------------------------------------------
// LSTM gate GEMV: g[r] = W1[r,:]·x + b1[r] + W2[r,:]·h + b2[r]
// ---------------------------------------------------------------------------
__global__ void gemv_gates(const float* __restrict__ W1, const float* __restrict__ x, int n1,
                           const float* __restrict__ W2, const float* __restrict__ h, int n2,
                           const float* __restrict__ b1, const float* __restrict__ b2,
                           float* __restrict__ g, int rows) {
    int wave = (blockIdx.x * blockDim.x + threadIdx.x) >> 5;
    int lane = threadIdx.x & 31;
    if (wave >= rows) return;
    float acc = 0.0f;
    {
        const float4* wr = (const float4*)(W1 + (size_t)wave * n1);
        const float4* xr = (const float4*)x;
        int nv = n1 >> 2;
        for (int i = lane; i < nv; i += 32) {
            float4 wv = wr[i], xv = xr[i];
            acc = fmaf(wv.x, xv.x, acc); acc = fmaf(wv.y, xv.y, acc);
            acc = fmaf(wv.z, xv.z, acc); acc = fmaf(wv.w, xv.w, acc);
        }
    }
    {
        const float4* wr = (const float4*)(W2 + (size_t)wave * n2);
        const float4* xr = (const float4*)h;
        int nv = n2 >> 2;
        for (int i = lane; i < nv; i += 32) {
            float4 wv = wr[i], xv = xr[i];
            acc = fmaf(wv.x, xv.x, acc); acc = fmaf(wv.y, xv.y, acc);
            acc = fmaf(wv.z, xv.z, acc); acc = fmaf(wv.w, xv.w, acc);
        }
    }
#pragma unroll
    for (int off = 16; off > 0; off >>= 1) acc += __shfl_down(acc, off, 32);
    if (lane == 0) g[wave] = acc + b1[wave] + b2[wave];
}

// ---------------------------------------------------------------------------
// Elementwise LSTM cell from packed gates (PyTorch order i,f,g,o), H = 1024
// ---------------------------------------------------------------------------
__global__ void lstm_elem(const float* __restrict__ g, const float* __restrict__ c_in,
                          float* __restrict__ h_out, float* __restrict__ c_out) {
    int j = blockIdx.x * blockDim.x + threadIdx.x;
    if (j >= H1) return;
    float ig = sigm(g[j]);
    float fg = sigm(g[H1 + j]);
    float gg = tanhf(g[2 * H1 + j]);
    float og = sigm(g[3 * H1 + j]);
    float c  = fg * c_in[j] + ig * gg;
    c_out[j] = c;
    h_out[j] = og * tanhf(c);
}

// ---------------------------------------------------------------------------
// vt = V(576x2048) @ lv_w.T(2048x512) with V_WMMA_F32_16X16X4_F32.
//
// One wave computes a 16x64 stripe: 4 accumulators sharing one A fragment
// (A reuse x4 -> 5 b64 loads feed 4 WMMAs per K-step of 4).
// Uniform scalar K-loop (lane offset folded into the load index) so EXEC
// stays all-1s through every WMMA, as the ISA requires.
//
// A frag (16x4 f32): lanes 0-15 -> M=lane, K={k0,k0+1}; lanes 16-31 -> K={k0+2,k0+3}.
// B frag (4x16 f32): lanes 0-15 -> N=lane, K={k0,k0+1}; lanes 16-31 -> K={k0+2,k0+3}.
// B[k][n] = lv_w[n*2048 + k]; the 4 B tiles sit at constant 128KB strides,
// which fit the 24-bit instruction offset (one shared address register).
// ---------------------------------------------------------------------------
__global__ void vt_wmma(const float* __restrict__ V, const float* __restrict__ lvw,
                        float* __restrict__ vt) {
    int wave = (blockIdx.x * blockDim.x + threadIdx.x) >> 5;   // 0..287
    int lane = threadIdx.x & 31;
    int tm = wave >> 3;           // 0..35  (row tile, 16 rows)
    int tg = wave & 7;            // 0..7   (col group, 64 cols)
    int half = lane >> 4;         // 0/1
    int l15  = lane & 15;

    const float2* Arow = (const float2*)(V   + (size_t)(tm * 16 + l15) * IMG);
    const float2* Brow = (const float2*)(lvw + (size_t)(tg * 64 + l15) * IMG);
    const int BSTEP = (16 * IMG) / 2;   // float2 stride between adjacent col tiles

    v8f c0 = {}, c1 = {}, c2 = {}, c3 = {};
#pragma unroll 2
    for (int k0 = 0; k0 < IMG; k0 += 4) {        // uniform trip count
        int k2 = (k0 >> 1) + half;               // float2 index incl. lane half
        float2 av  = Arow[k2];
        float2 b0v = Brow[k2];
        float2 b1v = Brow[k2 + BSTEP];
        float2 b2v = Brow[k2 + 2 * BSTEP];
        float2 b3v = Brow[k2 + 3 * BSTEP];
        v2f a;  a.x  = av.x;  a.y  = av.y;
        v2f b0; b0.x = b0v.x; b0.y = b0v.y;
        v2f b1; b1.x = b1v.x; b1.y = b1v.y;
        v2f b2; b2.x = b2v.x; b2.y = b2v.y;
        v2f b3; b3.x = b3v.x; b3.y = b3v.y;
        c0 = __builtin_amdgcn_wmma_f32_16x16x4_f32(false, a, false, b0, (short)0, c0, false, false);
        c1 = __builtin_amdgcn_wmma_f32_16x16x4_f32(false, a, false, b1, (short)0, c1, false, false);
        c2 = __builtin_amdgcn_wmma_f32_16x16x4_f32(false, a, false, b2, (short)0, c2, false, false);
        c3 = __builtin_amdgcn_wmma_f32_16x16x4_f32(false, a, false, b3, (short)0, c3, false, false);
    }
    // C/D layout: VGPR r -> M = r + half*8 (within tile), N = l15
    int mbase = tm * 16 + half * 8;
    int col   = tg * 64 + l15;
#pragma unroll
    for (int r = 0; r < 8; ++r) {
        float* row = vt + (size_t)(mbase + r) * MID + col;
        row[0]  = c0[r];
        row[16] = c1[r];
        row[32] = c2[r];
        row[48] = c3[r];
    }
}

// ---------------------------------------------------------------------------
// Attention logits: L[n] = att_b + sum_m tanh(vt[n][m] + lv_b[m] + h1t[m]) * att_w[m]
// (h1t already includes lh_b). One wave per region.
// ---------------------------------------------------------------------------
__global__ void att_logits(const float* __restrict__ vt, const float* __restrict__ lvb,
                           const float* __restrict__ h1t, const float* __restrict__ attw,
                           const float* __restrict__ attb, float* __restrict__ L) {
    int wave = (blockIdx.x * blockDim.x + threadIdx.x) >> 5;
    int lane = threadIdx.x & 31;
    if (wave >= N_REG) return;
    const float* vrow = vt + (size_t)wave * MID;
    float acc = 0.0f;
    for (int m = lane; m < MID; m += 32)
        acc = fmaf(tanhf(vrow[m] + lvb[m] + h1t[m]), attw[m], acc);
#pragma unroll
    for (int off = 16; off > 0; off >>= 1) acc += __shfl_down(acc, off, 32);
    if (lane == 0) L[wave] = acc + attb[0];
}

// ---------------------------------------------------------------------------
// Softmax over 576 regions, single block of 1024 threads (32 waves).
// ---------------------------------------------------------------------------
__global__ void softmax576(const float* __restrict__ L, float* __restrict__ a) {
    __shared__ float sm[32];
    int t = threadIdx.x;
    float x = (t < N_REG) ? L[t] : -INFINITY;

    float m = x;
#pragma unroll
    for (int off = 16; off > 0; off >>= 1) m = fmaxf(m, __shfl_down(m, off, 32));
    if ((t & 31) == 0) sm[t >> 5] = m;
    __syncthreads();
    if (t < 32) {
        float mm = sm[t];
#pragma unroll
        for (int off = 16; off > 0; off >>= 1) mm = fmaxf(mm, __shfl_down(mm, off, 32));
        if (t == 0) sm[0] = mm;
    }
    __syncthreads();
    float gmax = sm[0];
    __syncthreads();

    float e = (t < N_REG) ? __expf(x - gmax) : 0.0f;
    float s = e;
#pragma unroll
    for (int off = 16; off > 0; off >>= 1) s += __shfl_down(s, off, 32);
    if ((t & 31) == 0) sm[t >> 5] = s;
    __syncthreads();
    if (t < 32) {
        float ss = sm[t];
#pragma unroll
        for (int off = 16; off > 0; off >>= 1) ss += __shfl_down(ss, off, 32);
        if (t == 0) sm[0] = ss;
    }
    __syncthreads();
    if (t < N_REG) a[t] = e / sm[0];
}

// ---------------------------------------------------------------------------
// x2 = [ sum_n a[n]*V[n,:] (2048) | h1n (1024) ]
// ---------------------------------------------------------------------------
__global__ void build_x2(const float* __restrict__ V, const float* __restrict__ a,
                         const float* __restrict__ h1n, float* __restrict__ x2) {
    int j = blockIdx.x * blockDim.x + threadIdx.x;
    if (j >= IN2) return;
    if (j < IMG) {
        float acc = 0.0f;
        for (int n = 0; n < N_REG; ++n)
            acc = fmaf(a[n], V[(size_t)n * IMG + j], acc);
        x2[j] = acc;
    } else {
        x2[j] = h1n[j - IMG];
    }
}

// ---------------------------------------------------------------------------
extern "C" void kernel_launch(void* const* d_in, const int* in_sizes, int n_in,
                              void* d_out, int out_size, void* d_ws, size_t ws_size,
                              hipStream_t stream) {
    const float* V     = (const float*)d_in[0];
    const int*   w     = (const int*)  d_in[1];
    const float* h1    = (const float*)d_in[2];
    const float* c1    = (const float*)d_in[3];
    const float* h2    = (const float*)d_in[4];
    const float* c2    = (const float*)d_in[5];
    const float* emb   = (const float*)d_in[6];
    const float* W_ih1 = (const float*)d_in[7];
    const float* W_hh1 = (const float*)d_in[8];
    const float* b_ih1 = (const float*)d_in[9];
    const float* b_hh1 = (const float*)d_in[10];
    const float* W_ih2 = (const float*)d_in[11];
    const float* W_hh2 = (const float*)d_in[12];
    const float* b_ih2 = (const float*)d_in[13];
    const float* b_hh2 = (const float*)d_in[14];
    const float* lv_w  = (const float*)d_in[15];
    const float* lv_b  = (const float*)d_in[16];
    const float* lh_w  = (const float*)d_in[17];
    const float* lh_b  = (const float*)d_in[18];
    const float* att_w = (const float*)d_in[19];
    const float* att_b = (const float*)d_in[20];
    const float* lin_w = (const float*)d_in[21];
    const float* lin_b = (const float*)d_in[22];

    float* out = (float*)d_out;
    float* o    = out;                  // 32000
    float* h1n  = out + VOCAB;          // 1024
    float* c1n  = out + VOCAB + 1024;   // 1024
    float* h2n  = out + VOCAB + 2048;   // 1024
    float* c2n  = out + VOCAB + 3072;   // 1024

    float* ws  = (float*)d_ws;
    float* x1  = ws;               // 4096
    float* g1  = ws + 4096;        // 4096
    float* h1t = ws + 8192;        // 512
    float* L   = ws + 8704;        // 576
    float* att = ws + 9280;        // 576
    float* x2  = ws + 9856;        // 3072
    float* g2  = ws + 12928;       // 4096
    float* vt  = ws + 17024;       // 576*512 = 294912
    (void)in_sizes; (void)n_in; (void)out_size; (void)ws_size;

    // 1) x1 = [h2 | mean(V) | emb[w]]
    build_x1<<<IN1 / 256, 256, 0, stream>>>(V, h2, emb, w, x1);

    // 2) LSTM1 gates + cell
    gemv_gates<<<(4 * H1) / 8, 256, 0, stream>>>(W_ih1, x1, IN1, W_hh1, h1,
                                                 H1, b_ih1, b_hh1, g1, 4 * H1);
    lstm_elem<<<H1 / 256, 256, 0, stream>>>(g1, c1, h1n, c1n);

    // 3) h1t = lh_w @ h1n + lh_b
    gemv_bias<<<(MID + 7) / 8, 256, 0, stream>>>(lh_w, h1n, lh_b, h1t, MID, H1);

    // 4) vt = V @ lv_w.T   (WMMA f32 16x16x4; 36 x 8 waves, 16x64 per wave)
    vt_wmma<<<36, 256, 0, stream>>>(V, lv_w, vt);

    // 5) attention logits + softmax
    att_logits<<<(N_REG + 7) / 8, 256, 0, stream>>>(vt, lv_b, h1t, att_w, att_b, L);
    softmax576<<<1, 1024, 0, stream>>>(L, att);

    // 6) x2 = [attention-weighted V | h1n]
    build_x2<<<IN2 / 256, 256, 0, stream>>>(V, att, h1n, x2);

    // 7) LSTM2 gates + cell
    gemv_gates<<<(4 * H2) / 8, 256, 0, stream>>>(W_ih2, x2, IN2, W_hh2, h2,
                                                 H2, b_ih2, b_hh2, g2, 4 * H2);
    lstm_elem<<<H2 / 256, 256, 0, stream>>>(g2, c2, h2n, c2n);

    // 8) o = lin_w @ h2n + lin_b   (128 MB weight stream — the bandwidth hot spot)
    gemv_bias<<<VOCAB / 8, 256, 0, stream>>>(lin_w, h2n, lin_b, o, VOCAB, H2);
}